// GIB_GATConv_6794638262428
// MI455X (gfx1250) — compile-verified
//
#include <hip/hip_runtime.h>
#include <math.h>

typedef __attribute__((ext_vector_type(2))) float v2f;
typedef __attribute__((ext_vector_type(8))) float v8f;
typedef __attribute__((ext_vector_type(4))) unsigned v4u;
typedef __attribute__((ext_vector_type(8))) int v8i;
typedef __attribute__((ext_vector_type(4))) int v4i;

constexpr int NN   = 100000;   // nodes
constexpr int EE   = 1600000;  // edges
constexpr int FIN  = 128;      // IN
constexpr int FHID = 128;      // HID
constexpr int FOUT = 64;       // OUT
constexpr int NC   = 7;        // classes / clusters

#if __has_builtin(__builtin_amdgcn_tensor_load_to_lds) && __has_builtin(__builtin_amdgcn_s_wait_tensorcnt)
#define USE_TDM 1
#endif

// ---------------- workspace layout (float units) ----------------
constexpr size_t OFF_F0   = 0;                                  // N*HID  (h = x@W1)
constexpr size_t OFF_V    = OFF_F0 + (size_t)NN * FHID;         // N*HID  (v)
constexpr size_t OFF_LOG  = OFF_V + (size_t)NN * FHID;          // E      (edge logits)
constexpr size_t OFF_SS   = OFF_LOG + (size_t)EE;               // N      (h . a_src)
constexpr size_t OFF_SD   = OFF_SS + (size_t)NN;                // N      (h . a_dst)
constexpr size_t OFF_SMAX = OFF_SD + (size_t)NN;                // N      (segment max, ordered uint)
constexpr size_t OFF_SSUM = OFF_SMAX + (size_t)NN;              // N      (segment exp-sum)
constexpr size_t OFF_CSUM = OFF_SSUM + (size_t)NN;              // C*HID  (centroid accum)
constexpr size_t OFF_CNT  = OFF_CSUM + (size_t)NC * FHID;       // C
constexpr size_t OFF_CA   = ((OFF_CNT + NC + 3) / 4) * 4;       // C*HID  (centroids, 16B aligned)

// ---------------- helpers ----------------
__device__ __forceinline__ unsigned f2ord(float f) {
  unsigned u = __float_as_uint(f);
  return (u & 0x80000000u) ? ~u : (u | 0x80000000u);
}
__device__ __forceinline__ float ord2f(unsigned u) {
  return (u & 0x80000000u) ? __uint_as_float(u & 0x7fffffffu) : __uint_as_float(~u);
}
__device__ __forceinline__ float waveAllSum(float v) {
  #pragma unroll
  for (int off = 16; off >= 1; off >>= 1) v += __shfl_xor(v, off, 32);
  return v;
}

#ifdef USE_TDM
// Issue a TDM copy of `nElem` fp32 words from global `gsrc` into LDS `ldst`.
// D# per ISA ch.8: group0 = {count=1 | lds_addr | global_addr | type=2},
// group1 = 1-row tile of nElem 4-byte elements. Must be reached by one wave only.
// This toolchain's builtin takes 6 args (4 SGPR groups + extra group + cpol).
__device__ __forceinline__ void tdm_copy_to_lds(const float* gsrc, float* ldst, unsigned nElem) {
  unsigned long long ga = (unsigned long long)(uintptr_t)gsrc;
  unsigned la = (unsigned)(uintptr_t)ldst;  // low 32 bits of shared generic ptr = LDS byte offset
  v4u g0;
  g0[0] = 1u;                                              // count=1, user mode
  g0[1] = la;                                              // lds_addr
  g0[2] = (unsigned)ga;                                    // global_addr[31:0]
  g0[3] = ((unsigned)(ga >> 32) & 0x01FFFFFFu) | 0x80000000u; // addr[56:32] | type=2
  v8i g1;
  g1[0] = 0x00020000;                                      // data_size=2 (4B), no flags
  g1[1] = (int)((nElem & 0xFFFFu) << 16);                  // tensor_dim0[15:0]
  g1[2] = (int)(((nElem >> 16) & 0xFFFFu) | (1u << 16));   // tensor_dim0[31:16], tensor_dim1=1
  g1[3] = (int)((nElem & 0xFFFFu) << 16);                  // tile_dim0 = nElem
  g1[4] = 1;                                               // tile_dim1=1, tile_dim2=0
  g1[5] = (int)nElem;                                      // tensor_dim0_stride[31:0]
  g1[6] = 0;
  g1[7] = 0;
  v4i gz4 = {0, 0, 0, 0};
  v8i gz8 = {0, 0, 0, 0, 0, 0, 0, 0};
  __builtin_amdgcn_tensor_load_to_lds(g0, g1, gz4, gz4, gz8, 0);
  __builtin_amdgcn_s_wait_tensorcnt(0);
}
#endif

// ---------------- K0: init ----------------
__global__ void k_init(float* __restrict__ Z0, float* __restrict__ OUT0,
                       unsigned* __restrict__ SMAX, float* __restrict__ SSUM,
                       float* __restrict__ CSUM, float* __restrict__ CNT,
                       const float* __restrict__ b1, const float* __restrict__ b2) {
  int idx = blockIdx.x * blockDim.x + threadIdx.x;
  int stride = gridDim.x * blockDim.x;
  for (int i = idx; i < NN * FHID; i += stride) Z0[i] = b1[i & (FHID - 1)];
  for (int i = idx; i < NN * FOUT; i += stride) OUT0[i] = b2[i & (FOUT - 1)];
  for (int i = idx; i < NN; i += stride) { SMAX[i] = 0x007fffffu; SSUM[i] = 0.f; }
  for (int i = idx; i < NC * FHID; i += stride) CSUM[i] = 0.f;
  for (int i = idx; i < NC; i += stride) CNT[i] = 0.f;
}

// ---------------- K1: F0 = x @ W1  (fp32 WMMA, LDS-staged, 8 waves/block) ----------------
// Block: 256 thr = 8 waves; each wave owns one 16-row strip x all 8 col tiles.
// W1 (64KB) staged via TDM (or fallback), X strip (128 rows, 64KB) staged coalesced.
__global__ __launch_bounds__(256) void k_gemm1(const float* __restrict__ X,
                                               const float* __restrict__ W,
                                               float* __restrict__ F0) {
  __shared__ float lw[FHID * FHID];      // W1 row-major [k][n]
  __shared__ float xs[128 * FIN];        // 8 row-tiles of X
  const int tid = threadIdx.x;
#ifdef USE_TDM
  if (tid == 0) tdm_copy_to_lds(W, lw, FHID * FHID);
#else
  for (int i = tid; i < FHID * FHID / 4; i += 256)
    ((float4*)lw)[i] = ((const float4*)W)[i];
#endif
  const int rowBase = blockIdx.x * 128;
  for (int i = tid; i < 128 * (FIN / 4); i += 256) {
    int r = i >> 5;                         // FIN/4 = 32 float4 per row
    int gr = rowBase + r; if (gr >= NN) gr = NN - 1;
    ((float4*)xs)[i] = ((const float4*)(X + (size_t)gr * FIN))[i & 31];
  }
  __syncthreads();
  const int wave = tid >> 5, lane = tid & 31;
  const int rowTile = blockIdx.x * 8 + wave;
  if (rowTile >= NN / 16) return;
  const int lm = lane & 15;
  const int lh = lane >> 4;                 // lane half: K pair select / M+8 select
  const float* arow = xs + (wave * 16 + lm) * FIN;
  v8f acc[8] = {};
  for (int k0 = 0; k0 < FIN; k0 += 4) {
    const int ka = k0 + 2 * lh;
    v2f A = *(const v2f*)(arow + ka);       // ds_load_b64
    #pragma unroll
    for (int ct = 0; ct < 8; ++ct) {
      const int n = ct * 16 + lm;
      v2f B; B[0] = lw[ka * FHID + n]; B[1] = lw[(ka + 1) * FHID + n];
      acc[ct] = __builtin_amdgcn_wmma_f32_16x16x4_f32(false, A, false, B, (short)0,
                                                      acc[ct], false, false);
    }
  }
  float* dbase = F0 + (size_t)(rowTile * 16 + lh * 8) * FHID + lm;
  #pragma unroll
  for (int ct = 0; ct < 8; ++ct) {
    float* drow = dbase + ct * 16;
    #pragma unroll
    for (int r = 0; r < 8; ++r) drow[(size_t)r * FHID] = acc[ct][r];
  }
}

// ---------------- K2: per-node attention scores ----------------
__global__ __launch_bounds__(128) void k_scores(const float* __restrict__ F0,
                                                const float* __restrict__ a_src,
                                                const float* __restrict__ a_dst,
                                                float* __restrict__ SS, float* __restrict__ SD) {
  const int wave = threadIdx.x >> 5, lane = threadIdx.x & 31;
  const int node = blockIdx.x * 4 + wave;
  if (node >= NN) return;
  float4 h = ((const float4*)(F0 + (size_t)node * FHID))[lane];
  float4 as = ((const float4*)a_src)[lane];
  float4 ad = ((const float4*)a_dst)[lane];
  float ps = h.x * as.x + h.y * as.y + h.z * as.z + h.w * as.w;
  float pd = h.x * ad.x + h.y * ad.y + h.z * ad.z + h.w * ad.w;
  ps = waveAllSum(ps);
  pd = waveAllSum(pd);
  if (lane == 0) { SS[node] = ps; SD[node] = pd; }
}

// ---------------- K3: edge logits + segment max ----------------
__global__ void k_logits(const int* __restrict__ src, const int* __restrict__ dst,
                         const float* __restrict__ SS, const float* __restrict__ SD,
                         float* __restrict__ LOGv, unsigned* __restrict__ SMAX) {
  int e = blockIdx.x * blockDim.x + threadIdx.x;
  if (e >= EE) return;
  float l = SS[src[e]] + SD[dst[e]];
  l = l > 0.f ? l : 0.2f * l;  // leaky relu, slope 0.2
  LOGv[e] = l;
  atomicMax(&SMAX[dst[e]], f2ord(l));
}

// ---------------- K4: segment exp-sum ----------------
__global__ void k_expsum(const int* __restrict__ dst, const float* __restrict__ LOGv,
                         const unsigned* __restrict__ SMAX, float* __restrict__ SSUM) {
  int e = blockIdx.x * blockDim.x + threadIdx.x;
  if (e >= EE) return;
  int d = dst[e];
  atomicAdd(&SSUM[d], __expf(LOGv[e] - ord2f(SMAX[d])));
}

// ---------------- K5: Z0 += alpha * F0[src]  (Z0 pre-seeded with b1) ----------------
__global__ __launch_bounds__(256) void k_agg1(const int* __restrict__ src, const int* __restrict__ dst,
                                              const float* __restrict__ LOGv,
                                              const unsigned* __restrict__ SMAX,
                                              const float* __restrict__ SSUM,
                                              const float* __restrict__ F0, float* __restrict__ Z0) {
  const int wave = threadIdx.x >> 5, lane = threadIdx.x & 31;
  const int e = blockIdx.x * 8 + wave;
  if (e >= EE) return;
  const int s = src[e], d = dst[e];
  const float alpha = __expf(LOGv[e] - ord2f(SMAX[d])) / (SSUM[d] + 1e-16f);
  float4 h = ((const float4*)(F0 + (size_t)s * FHID))[lane];
  float* zr = Z0 + (size_t)d * FHID + lane * 4;
  atomicAdd(zr + 0, alpha * h.x);
  atomicAdd(zr + 1, alpha * h.y);
  atomicAdd(zr + 2, alpha * h.z);
  atomicAdd(zr + 3, alpha * h.w);
}

// ---------------- K6: class centroids (LDS hierarchical) ----------------
__global__ __launch_bounds__(256) void k_cent_accum(const float* __restrict__ Z0,
                                                    const int* __restrict__ Y,
                                                    float* __restrict__ CSUM, float* __restrict__ CNT) {
  __shared__ float ls[NC * FHID];
  __shared__ float lc[NC];
  for (int i = threadIdx.x; i < NC * FHID; i += 256) ls[i] = 0.f;
  if (threadIdx.x < NC) lc[threadIdx.x] = 0.f;
  __syncthreads();
  const int total = NN * FHID;
  for (int idx = blockIdx.x * 256 + threadIdx.x; idx < total; idx += gridDim.x * 256) {
    int node = idx >> 7;
    int dim = idx & (FHID - 1);
    int cls = Y[node];
    atomicAdd(&ls[cls * FHID + dim], Z0[idx]);
    if (dim == 0) atomicAdd(&lc[cls], 1.f);
  }
  __syncthreads();
  for (int i = threadIdx.x; i < NC * FHID; i += 256) atomicAdd(&CSUM[i], ls[i]);
  if (threadIdx.x < NC) atomicAdd(&CNT[threadIdx.x], lc[threadIdx.x]);
}

__global__ void k_cent_fin(const float* __restrict__ CSUM, const float* __restrict__ CNT,
                           float* __restrict__ CA) {
  int t = threadIdx.x;
  if (t < NC * FHID) CA[t] = CSUM[t] / (CNT[t >> 7] + 1e-12f);
}

// ---------------- K8: phi_Z, phi_X, S, v  (wave per node) ----------------
__global__ __launch_bounds__(128) void k_phi_v(const float* __restrict__ Z0,
                                               const float* __restrict__ Xin,
                                               const float* __restrict__ CA,
                                               const float* __restrict__ Cb,
                                               const float* __restrict__ Q,
                                               const int* __restrict__ Y,
                                               float* __restrict__ V, float* __restrict__ PHI) {
  const int wave = threadIdx.x >> 5, lane = threadIdx.x & 31;
  const int node = blockIdx.x * 4 + wave;
  if (node >= NN) return;
  float4 z = ((const float4*)(Z0 + (size_t)node * FHID))[lane];
  float4 xv = ((const float4*)(Xin + (size_t)node * FIN))[lane];

  float phiZ[NC], distZ[NC];
  float ssum = 0.f;
  #pragma unroll
  for (int k = 0; k < NC; ++k) {
    float4 c = ((const float4*)(CA + (size_t)k * FHID))[lane];
    float dx = z.x - c.x, dy = z.y - c.y, dz = z.z - c.z, dw = z.w - c.w;
    float p = waveAllSum(dx * dx + dy * dy + dz * dz + dw * dw);
    float d = sqrtf(p);
    distZ[k] = d;
    float sc = __expf(-d) + 1e-10f;
    phiZ[k] = sc;
    ssum += sc;
  }
  float inv = 1.f / ssum;
  #pragma unroll
  for (int k = 0; k < NC; ++k) phiZ[k] *= inv;

  // phi_X over input features -> S = sum_k log(phi_X_k)
  float slog = 0.f, sx = 0.f;
  #pragma unroll
  for (int k = 0; k < NC; ++k) {
    float4 c = ((const float4*)(Cb + (size_t)k * FIN))[lane];
    float dx = xv.x - c.x, dy = xv.y - c.y, dz = xv.z - c.z, dw = xv.w - c.w;
    float p = waveAllSum(dx * dx + dy * dy + dz * dz + dw * dw);
    float sc = __expf(-sqrtf(p)) + 1e-10f;
    sx += sc;
    slog += __logf(sc);
  }
  float S = slog - (float)NC * __logf(sx);

  const int yc = Y[node];
  float w[NC], Wsum = 0.f;
  #pragma unroll
  for (int k = 0; k < NC; ++k) {
    float diff = phiZ[k] * S - __logf(Q[yc * NC + k]);
    w[k] = phiZ[k] * diff;
    Wsum += w[k];
  }
  // v[d] = sum_k (w_k - Wsum*phiZ_k) * g_k[d],  g_k = -(z - c_k)/(dist_k + 1e-12)
  float vx = 0.f, vy = 0.f, vz = 0.f, vw = 0.f;
  #pragma unroll
  for (int k = 0; k < NC; ++k) {
    float4 c = ((const float4*)(CA + (size_t)k * FHID))[lane];
    float ck = w[k] - Wsum * phiZ[k];
    float sgn = -ck / (distZ[k] + 1e-12f);
    vx += sgn * (z.x - c.x);
    vy += sgn * (z.y - c.y);
    vz += sgn * (z.z - c.z);
    vw += sgn * (z.w - c.w);
  }
  float4 vout; vout.x = vx; vout.y = vy; vout.z = vz; vout.w = vw;
  ((float4*)(V + (size_t)node * FHID))[lane] = vout;
  if (lane == 0) {
    #pragma unroll
    for (int k = 0; k < NC; ++k) PHI[(size_t)node * NC + k] = phiZ[k];
  }
}

// ---------------- K9: h2 = elu(Z0) @ W2  (fp32 WMMA, ELU fused into LDS fill) ----------------
__global__ __launch_bounds__(256) void k_gemm2(const float* __restrict__ Z0,
                                               const float* __restrict__ W2,
                                               float* __restrict__ H2) {
  __shared__ float lw[FHID * FOUT];      // W2 row-major [k][n] (32KB)
  __shared__ float xs[128 * FHID];       // 8 row-tiles of elu(Z0) (64KB)
  const int tid = threadIdx.x;
#ifdef USE_TDM
  if (tid == 0) tdm_copy_to_lds(W2, lw, FHID * FOUT);
#else
  for (int i = tid; i < FHID * FOUT / 4; i += 256)
    ((float4*)lw)[i] = ((const float4*)W2)[i];
#endif
  const int rowBase = blockIdx.x * 128;
  for (int i = tid; i < 128 * (FHID / 4); i += 256) {
    int r = i >> 5;
    int gr = rowBase + r; if (gr >= NN) gr = NN - 1;
    float4 zv = ((const float4*)(Z0 + (size_t)gr * FHID))[i & 31];
    zv.x = zv.x > 0.f ? zv.x : (__expf(zv.x) - 1.f);   // elu (alpha=1)
    zv.y = zv.y > 0.f ? zv.y : (__expf(zv.y) - 1.f);
    zv.z = zv.z > 0.f ? zv.z : (__expf(zv.z) - 1.f);
    zv.w = zv.w > 0.f ? zv.w : (__expf(zv.w) - 1.f);
    ((float4*)xs)[i] = zv;
  }
  __syncthreads();
  const int wave = tid >> 5, lane = tid & 31;
  const int rowTile = blockIdx.x * 8 + wave;
  if (rowTile >= NN / 16) return;
  const int lm = lane & 15;
  const int lh = lane >> 4;
  const float* arow = xs + (wave * 16 + lm) * FHID;
  v8f acc[4] = {};
  for (int k0 = 0; k0 < FHID; k0 += 4) {
    const int ka = k0 + 2 * lh;
    v2f A = *(const v2f*)(arow + ka);
    #pragma unroll
    for (int ct = 0; ct < 4; ++ct) {
      const int n = ct * 16 + lm;
      v2f B; B[0] = lw[ka * FOUT + n]; B[1] = lw[(ka + 1) * FOUT + n];
      acc[ct] = __builtin_amdgcn_wmma_f32_16x16x4_f32(false, A, false, B, (short)0,
                                                      acc[ct], false, false);
    }
  }
  float* dbase = H2 + (size_t)(rowTile * 16 + lh * 8) * FOUT + lm;
  #pragma unroll
  for (int ct = 0; ct < 4; ++ct) {
    float* drow = dbase + ct * 16;
    #pragma unroll
    for (int r = 0; r < 8; ++r) drow[(size_t)r * FOUT] = acc[ct][r];
  }
}

// ---------------- K10: conv2 edge kernel (alpha, grad, B1, scatter) ----------------
__global__ __launch_bounds__(256) void k_conv2(const int* __restrict__ src, const int* __restrict__ dst,
                                               const float* __restrict__ LOGv,
                                               const unsigned* __restrict__ SMAX,
                                               const float* __restrict__ SSUM,
                                               const float* __restrict__ V,
                                               const float* __restrict__ F0,
                                               const float* __restrict__ H2,
                                               float* __restrict__ OUT0) {
  const int wave = threadIdx.x >> 5, lane = threadIdx.x & 31;
  const int e = blockIdx.x * 8 + wave;
  if (e >= EE) return;
  const int s = src[e], d = dst[e];
  const float alpha = __expf(LOGv[e] - ord2f(SMAX[d])) / (SSUM[d] + 1e-16f);
  float4 vv = ((const float4*)(V + (size_t)s * FHID))[lane];
  float4 ff = ((const float4*)(F0 + (size_t)d * FHID))[lane];
  float p = waveAllSum(vv.x * ff.x + vv.y * ff.y + vv.z * ff.z + vv.w * ff.w);
  float grad = p * (1.0f / (float)NN);
  float b = alpha - grad;
  b = b / (fabsf(b) + 1e-16f);                     // sign-normalize (|B1| detached)
  float2 h2 = ((const float2*)(H2 + (size_t)s * FOUT))[lane];
  atomicAdd(&OUT0[(size_t)d * FOUT + lane * 2 + 0], b * h2.x);
  atomicAdd(&OUT0[(size_t)d * FOUT + lane * 2 + 1], b * h2.y);
}

// ---------------- launcher ----------------
extern "C" void kernel_launch(void* const* d_in, const int* in_sizes, int n_in,
                              void* d_out, int out_size, void* d_ws, size_t ws_size,
                              hipStream_t stream) {
  (void)in_sizes; (void)n_in; (void)out_size; (void)ws_size;
  const float* X     = (const float*)d_in[0];
  const int*   EI    = (const int*)d_in[1];
  const int*   Y     = (const int*)d_in[2];
  const float* Cb    = (const float*)d_in[3];
  const float* Q     = (const float*)d_in[4];
  const float* W1    = (const float*)d_in[5];
  const float* a_src = (const float*)d_in[6];
  const float* a_dst = (const float*)d_in[7];
  const float* b1    = (const float*)d_in[8];
  const float* W2    = (const float*)d_in[9];
  const float* b2    = (const float*)d_in[10];
  const int* src = EI;
  const int* dst = EI + EE;

  float* OUT0 = (float*)d_out;                 // [N, OUT]
  float* H2   = OUT0 + (size_t)NN * FOUT;      // [N, OUT]
  float* Z0   = H2 + (size_t)NN * FOUT;        // [N, HID]
  float* PHI  = Z0 + (size_t)NN * FHID;        // [N, C]

  float*    ws   = (float*)d_ws;
  float*    F0   = ws + OFF_F0;
  float*    V    = ws + OFF_V;
  float*    LOGv = ws + OFF_LOG;
  float*    SS   = ws + OFF_SS;
  float*    SD   = ws + OFF_SD;
  unsigned* SMAX = (unsigned*)(ws + OFF_SMAX);
  float*    SSUM = ws + OFF_SSUM;
  float*    CSUM = ws + OFF_CSUM;
  float*    CNT  = ws + OFF_CNT;
  float*    CA   = ws + OFF_CA;

  const int rowTiles = NN / 16;                          // 6250
  k_init<<<2048, 256, 0, stream>>>(Z0, OUT0, SMAX, SSUM, CSUM, CNT, b1, b2);
  k_gemm1<<<(rowTiles + 7) / 8, 256, 0, stream>>>(X, W1, F0);
  k_scores<<<(NN + 3) / 4, 128, 0, stream>>>(F0, a_src, a_dst, SS, SD);
  k_logits<<<(EE + 255) / 256, 256, 0, stream>>>(src, dst, SS, SD, LOGv, SMAX);
  k_expsum<<<(EE + 255) / 256, 256, 0, stream>>>(dst, LOGv, SMAX, SSUM);
  k_agg1<<<(EE + 7) / 8, 256, 0, stream>>>(src, dst, LOGv, SMAX, SSUM, F0, Z0);
  k_cent_accum<<<1024, 256, 0, stream>>>(Z0, Y, CSUM, CNT);
  k_cent_fin<<<1, NC * FHID, 0, stream>>>(CSUM, CNT, CA);
  k_phi_v<<<(NN + 3) / 4, 128, 0, stream>>>(Z0, X, CA, Cb, Q, Y, V, PHI);
  k_gemm2<<<(rowTiles + 7) / 8, 256, 0, stream>>>(Z0, W2, H2);
  k_conv2<<<(EE + 7) / 8, 256, 0, stream>>>(src, dst, LOGv, SMAX, SSUM, V, F0, H2, OUT0);
}